// LinkPredictorGlobalModel_79405355368558
// MI455X (gfx1250) — compile-verified
//
#include <hip/hip_runtime.h>

// ---------------- problem constants (match reference) ----------------
#define NNODES 65536
#define NEDGES (1 << 20)
#define HCH    128
#define CIN    4
#define NBATCH 64
#define LLEN   1024   // nodes per graph
#define LP     512    // after maxpool2

typedef __attribute__((ext_vector_type(2))) float v2f;
typedef __attribute__((ext_vector_type(8))) float v8f;

__device__ __forceinline__ v8f wmma_f32_k4(v2f a, v2f b, v8f c) {
    // V_WMMA_F32_16X16X4_F32 : D = A(16x4,f32) * B(4x16,f32) + C(16x16,f32)
    return __builtin_amdgcn_wmma_f32_16x16x4_f32(
        /*neg_a=*/false, a, /*neg_b=*/false, b,
        /*c_mod=*/(short)0, c, /*reuse_a=*/false, /*reuse_b=*/false);
}

// ---------------- utility: zero float4-sized region ----------------
__global__ void zero_kernel(float* __restrict__ p, int n4) {
    int i = blockIdx.x * blockDim.x + threadIdx.x;
    if (i < n4) ((float4*)p)[i] = make_float4(0.f, 0.f, 0.f, 0.f);
}

// ---------------- in-degree histogram: cnt[dst]++ per edge (int, cheap) ----------------
__global__ void count_kernel(const int* __restrict__ ei, int* __restrict__ cnt) {
    int e = blockIdx.x * blockDim.x + threadIdx.x;   // grid == NEDGES
    atomicAdd(&cnt[ei[NEDGES + e]], 1);
}

// dinv[i] = rsqrt(1 + indeg[i])
__global__ void dinv_kernel(const int* __restrict__ cnt, float* __restrict__ dinv) {
    int i = blockIdx.x * blockDim.x + threadIdx.x;   // grid == NNODES
    dinv[i] = rsqrtf(1.0f + (float)cnt[i]);
}

// ---------------- hierarchical exclusive prefix sum over cnt[N] -> rowptr ----------------
// pass 1: per-block (256 elems) exclusive scan in LDS; rowptr[i] = local exclusive, bsum[b] = block total
__global__ void scan_block_kernel(const int* __restrict__ cnt, int* __restrict__ rowptr,
                                  int* __restrict__ bsum) {
    __shared__ int s[256];
    const int t = threadIdx.x;
    const int i = blockIdx.x * 256 + t;
    const int v = cnt[i];
    s[t] = v;
    __syncthreads();
#pragma unroll
    for (int off = 1; off < 256; off <<= 1) {
        int add = (t >= off) ? s[t - off] : 0;
        __syncthreads();
        s[t] += add;
        __syncthreads();
    }
    rowptr[i] = s[t] - v;                    // exclusive within block
    if (t == 255) bsum[blockIdx.x] = s[255]; // block total
}

// pass 2: single block exclusive scan of the 256 block sums (in place)
__global__ void scan_sums_kernel(int* __restrict__ bsum) {
    __shared__ int s[256];
    const int t = threadIdx.x;
    const int v = bsum[t];
    s[t] = v;
    __syncthreads();
#pragma unroll
    for (int off = 1; off < 256; off <<= 1) {
        int add = (t >= off) ? s[t - off] : 0;
        __syncthreads();
        s[t] += add;
        __syncthreads();
    }
    bsum[t] = s[t] - v;
}

// pass 3: add block offsets; also initialize fill cursors and rowptr[N]
__global__ void scan_add_kernel(int* __restrict__ rowptr, int* __restrict__ cursor,
                                const int* __restrict__ bsum) {
    const int i = blockIdx.x * 256 + threadIdx.x;
    const int r = rowptr[i] + bsum[blockIdx.x];
    rowptr[i] = r;
    cursor[i] = r;
    if (i == 0) rowptr[NNODES] = NEDGES;
}

// ---------------- CSR fill: bucket source ids by destination ----------------
__global__ void fill_kernel(const int* __restrict__ ei, int* __restrict__ cursor,
                            int* __restrict__ csrsrc) {
    int e = blockIdx.x * blockDim.x + threadIdx.x;   // grid == NEDGES
    const int src = ei[e];
    const int dst = ei[NEDGES + e];
    const int pos = atomicAdd(&cursor[dst], 1);
    csrsrc[pos] = src;
}

// ---------------- GEMM via WMMA f32 16x16x4: C[M,128] = (A[M,K] @ B[K,128]) * dinv[row] ----
// grid.x = M/16 (row tiles); block = 256 = 8 waves; wave w -> 16-col tile
__global__ void gemm_wmma_kernel(const float* __restrict__ A,
                                 const float* __restrict__ Bm,
                                 const float* __restrict__ dinv,
                                 float* __restrict__ C, int K) {
    const int lane = threadIdx.x & 31;
    const int wave = threadIdx.x >> 5;
    const int hi   = lane >> 4;       // 0: lanes 0-15, 1: lanes 16-31
    const int lo   = lane & 15;
    const int rowBase = blockIdx.x * 16;
    const int colBase = wave * 16;

    v8f acc = {0.f, 0.f, 0.f, 0.f, 0.f, 0.f, 0.f, 0.f};
    const float* arow = A + (size_t)(rowBase + lo) * K;

    for (int k0 = 0; k0 < K; k0 += 4) {
        const int keff = k0 + 2 * hi;
        v2f a, b;
        a.x = arow[keff];
        a.y = arow[keff + 1];
        b.x = Bm[(size_t)keff * HCH + colBase + lo];
        b.y = Bm[(size_t)(keff + 1) * HCH + colBase + lo];
        acc = wmma_f32_k4(a, b, acc);
    }

#pragma unroll
    for (int r = 0; r < 8; ++r) {
        const int row = rowBase + r + 8 * hi;        // C/D layout: M = r + 8*(lane>=16)
        C[(size_t)row * HCH + colBase + lo] = acc[r] * dinv[row];
    }
}

// ---------------- fused CSR gather + self-loop + bias + relu ----------------
// out[n] = relu( dinv[n] * ( sum_{e in row(n)} linS[csr_src[e]] + linS[n] ) + bias )
// one wave per node (32 lanes = 32 float4 column groups) -> uniform trip count per wave.
// Edge loop unrolled 4-wide so 4 independent index loads + 4 independent b128 gathers
// are in flight per iteration (MLP inside the wave, on top of wave-level switching).
__global__ void gather_finish_kernel(const int* __restrict__ rowptr,
                                     const int* __restrict__ csrsrc,
                                     const float* __restrict__ linS,
                                     const float* __restrict__ dinv,
                                     const float* __restrict__ bias,
                                     float* __restrict__ out) {
    const int idx = blockIdx.x * blockDim.x + threadIdx.x;   // grid == NNODES*32
    const int n = idx >> 5;
    const int g = idx & 31;
    const int s0 = rowptr[n];
    const int s1 = rowptr[n + 1];
    const float4* __restrict__ lin4 = (const float4*)linS;

    float4 acc = lin4[n * 32 + g];                           // self-loop term (pre-scaled)
    int e = s0;
    for (; e + 4 <= s1; e += 4) {
        const int i0 = csrsrc[e + 0];
        const int i1 = csrsrc[e + 1];
        const int i2 = csrsrc[e + 2];
        const int i3 = csrsrc[e + 3];
        const float4 v0 = lin4[i0 * 32 + g];
        const float4 v1 = lin4[i1 * 32 + g];
        const float4 v2 = lin4[i2 * 32 + g];
        const float4 v3 = lin4[i3 * 32 + g];
        acc.x += v0.x + v1.x + v2.x + v3.x;
        acc.y += v0.y + v1.y + v2.y + v3.y;
        acc.z += v0.z + v1.z + v2.z + v3.z;
        acc.w += v0.w + v1.w + v2.w + v3.w;
    }
    for (; e < s1; ++e) {
        const float4 v = lin4[csrsrc[e] * 32 + g];
        acc.x += v.x; acc.y += v.y; acc.z += v.z; acc.w += v.w;
    }

    const float d = dinv[n];
    const float4 bv = ((const float4*)bias)[g];
    float4 r;
    r.x = fmaxf(acc.x * d + bv.x, 0.f);
    r.y = fmaxf(acc.y * d + bv.y, 0.f);
    r.z = fmaxf(acc.z * d + bv.z, 0.f);
    r.w = fmaxf(acc.w * d + bv.w, 0.f);
    ((float4*)out)[idx] = r;
}

// ---------------- conv1 (C_IN=4 -> 128, k=3, pad=1) + relu + maxpool2 ----------------
// one thread per (b, oc, l2); pooled layout [b][oc][l2], grid = 64*128*512/256
__global__ void conv1_pool_kernel(const float* __restrict__ x,
                                  const float* __restrict__ cW1,
                                  const float* __restrict__ cb1,
                                  float* __restrict__ pooled) {
    const int idx = blockIdx.x * blockDim.x + threadIdx.x;
    const int l2 = idx & (LP - 1);
    const int oc = (idx >> 9) & (HCH - 1);
    const int b  = idx >> 16;
    const int l0 = 2 * l2;

    float4 xv[4];
#pragma unroll
    for (int p = 0; p < 4; ++p) {
        const int pos = l0 - 1 + p;
        xv[p] = (pos >= 0 && pos < LLEN) ? ((const float4*)x)[b * LLEN + pos]
                                         : make_float4(0.f, 0.f, 0.f, 0.f);
    }
    const float* w = cW1 + oc * (CIN * 3);
    const float bias = cb1[oc];
    float y0 = bias, y1 = bias;
#pragma unroll
    for (int ic = 0; ic < CIN; ++ic) {
        const float w0 = w[ic * 3 + 0], w1 = w[ic * 3 + 1], w2 = w[ic * 3 + 2];
        const float xm1 = ((const float*)&xv[0])[ic];
        const float x0  = ((const float*)&xv[1])[ic];
        const float x1p = ((const float*)&xv[2])[ic];
        const float x2p = ((const float*)&xv[3])[ic];
        y0 += w0 * xm1 + w1 * x0 + w2 * x1p;
        y1 += w0 * x0 + w1 * x1p + w2 * x2p;
    }
    y0 = fmaxf(y0, 0.f);
    y1 = fmaxf(y1, 0.f);
    pooled[idx] = fmaxf(y0, y1);
}

// ---------------- conv2 (128 -> 128, k=3, pad=1) + relu + global max, via WMMA ----------------
// k=3 conv expressed as 3 shifted GEMMs (K=128 each). grid.x = NBATCH*32 (L tiles);
// block = 256 = 8 waves; wave w -> oc tile [16w, 16w+16).
__global__ void conv2_max_kernel(const float* __restrict__ pooled,
                                 const float* __restrict__ cW2,
                                 const float* __restrict__ cb2,
                                 float* __restrict__ reps) {
    const int lane = threadIdx.x & 31;
    const int wave = threadIdx.x >> 5;
    const int hi = lane >> 4;
    const int lo = lane & 15;
    const int lt = blockIdx.x & 31;      // L tile index (512/16)
    const int b  = blockIdx.x >> 5;      // batch
    const int ocBase = wave * 16;
    const int Lbase  = lt * 16;

    const float* pb = pooled + (size_t)b * HCH * LP;
    v8f acc = {0.f, 0.f, 0.f, 0.f, 0.f, 0.f, 0.f, 0.f};

#pragma unroll
    for (int kk = 0; kk < 3; ++kk) {
        const int pos = Lbase + lo + kk - 1;            // A-fragment row position
        const bool ok = (pos >= 0) && (pos < LP);
        for (int k0 = 0; k0 < HCH; k0 += 4) {
            const int ic = k0 + 2 * hi;
            v2f a, w;
            a.x = ok ? pb[ic * LP + pos] : 0.f;
            a.y = ok ? pb[(ic + 1) * LP + pos] : 0.f;
            const float* wp = cW2 + (size_t)(ocBase + lo) * (HCH * 3) + ic * 3 + kk;
            w.x = wp[0];
            w.y = wp[3];
            acc = wmma_f32_k4(a, w, acc);
        }
    }

    const float bias = cb2[ocBase + lo];
    float m = 0.f;                                      // relu => values >= 0
#pragma unroll
    for (int r = 0; r < 8; ++r) m = fmaxf(m, fmaxf(acc[r] + bias, 0.f));
    m = fmaxf(m, __shfl_xor(m, 16, 32));                // merge M=0..7 with M=8..15
    if (lane < 16) {
        // m >= 0, so float ordering == unsigned-bit ordering; reps pre-zeroed
        atomicMax((unsigned int*)&reps[b * HCH + ocBase + lane], __float_as_uint(m));
    }
}

// ---------------- launcher ----------------
extern "C" void kernel_launch(void* const* d_in, const int* in_sizes, int n_in,
                              void* d_out, int out_size, void* d_ws, size_t ws_size,
                              hipStream_t stream) {
    (void)in_sizes; (void)n_in; (void)out_size; (void)ws_size;

    const float* x   = (const float*)d_in[0];
    const float* W1  = (const float*)d_in[1];
    const float* b1  = (const float*)d_in[2];
    const float* W2  = (const float*)d_in[3];
    const float* b2  = (const float*)d_in[4];
    const float* cW1 = (const float*)d_in[5];
    const float* cb1 = (const float*)d_in[6];
    const float* cW2 = (const float*)d_in[7];
    const float* cb2 = (const float*)d_in[8];
    const int*   ei  = (const int*)d_in[9];
    // d_in[10] = batch: unused (equal-size graphs -> dense batch is a reshape)

    float* outH = (float*)d_out;                         // [65536,128]
    float* reps = outH + (size_t)NNODES * HCH;           // [64,128]

    // ---- workspace layout (all 4-byte units) ----
    float* ws     = (float*)d_ws;
    float* dinv   = ws;                                  // N floats
    int*   cnt    = (int*)(ws + NNODES);                 // N
    int*   rowptr = cnt + NNODES;                        // N+1
    int*   cursor = rowptr + NNODES + 1;                 // N
    int*   csrsrc = cursor + NNODES;                     // E
    int*   bsum   = csrsrc + NEDGES;                     // 256
    // meta total = 4*N + 1 + E + 256 = 1,310,977 units; round to 1,310,980 (16B aligned)
    float* bufA   = ws + 1310980;                        // N*128 (linS)
    float* bufB   = bufA + (size_t)NNODES * HCH;         // N*128 (h / pooled)
    float* pooled = bufB;                                // 64*128*512 shares bufB (conv runs first)

    // zero init (harness poisons d_out/d_ws)
    zero_kernel<<<(NBATCH * HCH / 4 + 255) / 256, 256, 0, stream>>>(reps, NBATCH * HCH / 4);
    zero_kernel<<<(NNODES / 4 + 255) / 256, 256, 0, stream>>>((float*)cnt, NNODES / 4);

    // ---- conv branch (runs first; pooled shares bufB) ----
    conv1_pool_kernel<<<NBATCH * HCH * LP / 256, 256, 0, stream>>>(x, cW1, cb1, pooled);
    conv2_max_kernel<<<NBATCH * 32, 256, 0, stream>>>(pooled, cW2, cb2, reps);

    // ---- CSR build (once; shared by both GCN layers) ----
    count_kernel<<<NEDGES / 256, 256, 0, stream>>>(ei, cnt);
    dinv_kernel<<<NNODES / 256, 256, 0, stream>>>(cnt, dinv);
    scan_block_kernel<<<NNODES / 256, 256, 0, stream>>>(cnt, rowptr, bsum);
    scan_sums_kernel<<<1, 256, 0, stream>>>(bsum);
    scan_add_kernel<<<NNODES / 256, 256, 0, stream>>>(rowptr, cursor, bsum);
    fill_kernel<<<NEDGES / 256, 256, 0, stream>>>(ei, cursor, csrsrc);

    // ---- GCN layer 1: linS1 = (x@W1)*dinv ; fused gather+self+bias+relu -> bufB ----
    gemm_wmma_kernel<<<NNODES / 16, 256, 0, stream>>>(x, W1, dinv, bufA, CIN);
    gather_finish_kernel<<<NNODES * 32 / 256, 256, 0, stream>>>(rowptr, csrsrc, bufA, dinv, b1, bufB);

    // ---- GCN layer 2: linS2 = (h@W2)*dinv ; fused gather -> d_out ----
    gemm_wmma_kernel<<<NNODES / 16, 256, 0, stream>>>(bufB, W2, dinv, bufA, HCH);
    gather_finish_kernel<<<NNODES * 32 / 256, 256, 0, stream>>>(rowptr, csrsrc, bufA, dinv, b2, outH);
}